// GINEncoder_39831526703451
// MI455X (gfx1250) — compile-verified
//
#include <hip/hip_runtime.h>
#include <hip/hip_bf16.h>

// GIN encoder for MI455X (gfx1250, wave32).
// h lives in d_out (N*128 f32). d_ws = agg (N*128 f32, doubles as the pre-BN z
// buffer, rewritten in place by the MLP kernel) + 512 floats BN scratch.

typedef __attribute__((ext_vector_type(2))) float v2f;
typedef __attribute__((ext_vector_type(8))) float v8f;

#define H   128
#define H2  256
#define MT  64          // rows per MLP block (4 row tiles of 16)
#define ZS_STRIDE 130   // 130 % 64 == 2 -> conflict-free row walks
#define TS_STRIDE 258   // 258 % 64 == 2

// ---------------- embedding: h = x @ Wemb + b ; agg = h ----------------
__global__ void gin_embed(const float* __restrict__ x, const float* __restrict__ Wemb,
                          const float* __restrict__ bemb, float* __restrict__ h,
                          float* __restrict__ agg, long total, int Kin) {
    long tid = (long)blockIdx.x * blockDim.x + threadIdx.x;
    if (tid >= total) return;
    int n = (int)(tid >> 7);
    int j = (int)(tid & (H - 1));
    float s = bemb[j];
    for (int k = 0; k < Kin; ++k)
        s = fmaf(x[(long)n * Kin + k], Wemb[(long)k * H + j], s);
    h[tid]   = s;
    agg[tid] = s;
}

// ---------------- scatter: agg[dst] += h[src] (agg preset to h) --------
__global__ void gin_scatter(const float* __restrict__ h, const int* __restrict__ src,
                            const int* __restrict__ dst, float* __restrict__ agg,
                            long total) {
    long tid = (long)blockIdx.x * blockDim.x + threadIdx.x;
    if (tid >= total) return;
    int e = (int)(tid >> 5);          // 32 threads per edge
    int f = ((int)tid & 31) * 4;      // 4 floats per thread
    int s = src[e];
    int d = dst[e];
    const float4 v = *reinterpret_cast<const float4*>(h + (long)s * H + f);
    float* p = agg + (long)d * H + f;
    atomicAdd(p + 0, v.x);
    atomicAdd(p + 1, v.y);
    atomicAdd(p + 2, v.z);
    atomicAdd(p + 3, v.w);
}

// ---------------- fused MLP: z = relu(agg@W1+b1)@W2+b2 (in place) ------
// 8 wave32 per block, 64-row node tile per block (4 row tiles).
// Each B fragment (weights) is loaded once per k-step and reused across the
// 4 row tiles -> 4x less L2 weight traffic + 8 independent WMMA chains.
__global__ __launch_bounds__(256) void gin_mlp(
        float* __restrict__ agg,
        const float* __restrict__ W1, const float* __restrict__ b1,
        const float* __restrict__ W2, const float* __restrict__ b2,
        float* __restrict__ colsum, float* __restrict__ colsq, long N) {
    __shared__ float Zs[MT * ZS_STRIDE];   // input tile 64 x 128
    __shared__ float Ts[MT * TS_STRIDE];   // intermediate tile 64 x 256

    const int t    = threadIdx.x;
    const int lane = t & 31;
    const int wave = t >> 5;
    const long base = (long)blockIdx.x * MT;

    // stage input tile (coalesced, tail rows -> 0)
    for (int i = t; i < MT * H; i += 256) {
        int r = i >> 7, c = i & (H - 1);
        long row = base + r;
        Zs[r * ZS_STRIDE + c] = (row < N) ? agg[row * H + c] : 0.0f;
    }
    __syncthreads();

    // fragment coordinates per ISA 7.12.2:
    //   A 16x4 f32: lane L holds A[M=L&15][k+ksel], A[M=L&15][k+ksel+1]
    //   B 4x16 f32: lane L holds B[k+ksel][N=L&15], B[k+ksel+1][N=L&15]
    //   C/D 16x16:  VGPR j, lanes 0-15 -> M=j, lanes 16-31 -> M=j+8
    const int mrow  = lane & 15;
    const int ncol  = lane & 15;
    const int ksel  = (lane >> 4) << 1;   // 0 or 2
    const int rbase = (lane >> 4) << 3;   // 0 or 8

    // ---- GEMM1: 128 -> 256; wave owns cols [cb0,cb0+32), all 4 row tiles
    const int cb0 = wave * 32;
    const int cb1 = cb0 + 16;
    v8f accA[4], accB[4];
    {
        float bv0 = b1[cb0 + ncol], bv1 = b1[cb1 + ncol];
#pragma unroll
        for (int rt = 0; rt < 4; ++rt)
#pragma unroll
            for (int j = 0; j < 8; ++j) { accA[rt][j] = bv0; accB[rt][j] = bv1; }
    }
#pragma unroll 4
    for (int k = 0; k < H; k += 4) {
        const int kr = k + ksel;
        v2f bA; bA.x = W1[(long)kr * H2 + cb0 + ncol];
                bA.y = W1[(long)(kr + 1) * H2 + cb0 + ncol];
        v2f bB; bB.x = W1[(long)kr * H2 + cb1 + ncol];
                bB.y = W1[(long)(kr + 1) * H2 + cb1 + ncol];
#pragma unroll
        for (int rt = 0; rt < 4; ++rt) {
            v2f a; a.x = Zs[(rt * 16 + mrow) * ZS_STRIDE + kr];
                   a.y = Zs[(rt * 16 + mrow) * ZS_STRIDE + kr + 1];
            accA[rt] = __builtin_amdgcn_wmma_f32_16x16x4_f32(false, a, false, bA,
                                                             (short)0, accA[rt], false, false);
            accB[rt] = __builtin_amdgcn_wmma_f32_16x16x4_f32(false, a, false, bB,
                                                             (short)0, accB[rt], false, false);
        }
    }
    // ReLU -> LDS intermediate
#pragma unroll
    for (int rt = 0; rt < 4; ++rt)
#pragma unroll
        for (int j = 0; j < 8; ++j) {
            Ts[(rt * 16 + rbase + j) * TS_STRIDE + cb0 + ncol] = fmaxf(accA[rt][j], 0.0f);
            Ts[(rt * 16 + rbase + j) * TS_STRIDE + cb1 + ncol] = fmaxf(accB[rt][j], 0.0f);
        }
    __syncthreads();

    // ---- GEMM2: 256 -> 128; wave owns cols [cb2,cb2+16), all 4 row tiles
    const int cb2 = wave * 16;
    v8f acc2[4];
    {
        float bv = b2[cb2 + ncol];
#pragma unroll
        for (int rt = 0; rt < 4; ++rt)
#pragma unroll
            for (int j = 0; j < 8; ++j) acc2[rt][j] = bv;
    }
#pragma unroll 4
    for (int k = 0; k < H2; k += 4) {
        const int kr = k + ksel;
        v2f b; b.x = W2[(long)kr * H + cb2 + ncol];
               b.y = W2[(long)(kr + 1) * H + cb2 + ncol];
#pragma unroll
        for (int rt = 0; rt < 4; ++rt) {
            v2f a; a.x = Ts[(rt * 16 + mrow) * TS_STRIDE + kr];
                   a.y = Ts[(rt * 16 + mrow) * TS_STRIDE + kr + 1];
            acc2[rt] = __builtin_amdgcn_wmma_f32_16x16x4_f32(false, a, false, b,
                                                             (short)0, acc2[rt], false, false);
        }
    }

    // write z back in place; accumulate per-column BN partial sums
    float psum = 0.0f, psq = 0.0f;
#pragma unroll
    for (int rt = 0; rt < 4; ++rt)
#pragma unroll
        for (int j = 0; j < 8; ++j) {
            long row = base + rt * 16 + rbase + j;
            if (row < N) {
                float v = acc2[rt][j];
                agg[row * H + cb2 + ncol] = v;
                psum += v;
                psq  = fmaf(v, v, psq);
            }
        }
    // fold lane L and L+16 (same column, other rows)
    psum += __shfl_xor(psum, 16, 32);
    psq  += __shfl_xor(psq, 16, 32);
    if (lane < 16) {
        atomicAdd(&colsum[cb2 + ncol], psum);
        atomicAdd(&colsq [cb2 + ncol], psq);
    }
}

// ---------------- BN stats -> scale/shift ------------------------------
__global__ void gin_bnstat(const float* __restrict__ colsum, const float* __restrict__ colsq,
                           const float* __restrict__ gamma, const float* __restrict__ beta,
                           float* __restrict__ scale, float* __restrict__ shift, float invN) {
    int c = threadIdx.x;
    float mean = colsum[c] * invN;
    float var  = colsq[c] * invN - mean * mean;   // biased, matches jnp.var
    float sc   = gamma[c] * rsqrtf(var + 1e-5f);
    scale[c] = sc;
    shift[c] = beta[c] - mean * sc;
}

// ---------------- apply BN + ReLU; h = result; agg = h -----------------
__global__ void gin_apply(const float* __restrict__ scale, const float* __restrict__ shift,
                          float* __restrict__ z /* == agg */, float* __restrict__ h,
                          long total) {
    long i = (long)blockIdx.x * blockDim.x + threadIdx.x;
    if (i >= total) return;
    int c = (int)(i & (H - 1));
    float v = fmaxf(fmaf(z[i], scale[c], shift[c]), 0.0f);
    h[i] = v;
    z[i] = v;   // re-seed agg = h for next layer's scatter
}

__global__ void gin_zero(float* __restrict__ p, int n) {
    int i = blockIdx.x * blockDim.x + threadIdx.x;
    if (i < n) p[i] = 0.0f;
}

extern "C" void kernel_launch(void* const* d_in, const int* in_sizes, int n_in,
                              void* d_out, int out_size, void* d_ws, size_t ws_size,
                              hipStream_t stream) {
    const float* x     = (const float*)d_in[0];
    const int*   ei    = (const int*)  d_in[1];
    const float* embW  = (const float*)d_in[2];
    const float* embB  = (const float*)d_in[3];
    const float* W1    = (const float*)d_in[4];
    const float* b1    = (const float*)d_in[5];
    const float* W2    = (const float*)d_in[6];
    const float* b2    = (const float*)d_in[7];
    const float* gamma = (const float*)d_in[8];
    const float* beta  = (const float*)d_in[9];

    const int Hh  = in_sizes[3];              // 128
    const int Kin = in_sizes[2] / Hh;         // 32
    const int N   = in_sizes[0] / Kin;        // 100000
    const int E   = in_sizes[1] / 2;          // 400000
    const int L   = in_sizes[5] / (2 * Hh);   // 4

    float* h      = (float*)d_out;
    float* agg    = (float*)d_ws;             // N*H f32, doubles as z buffer
    float* colsum = agg + (size_t)N * Hh;
    float* colsq  = colsum + Hh;
    float* scale  = colsq + Hh;
    float* shift  = scale + Hh;

    const long totNH = (long)N * Hh;
    const int  blkNH = (int)((totNH + 255) / 256);
    const long totEf = (long)E * 32;          // 32 threads per edge
    const int  blkE  = (int)((totEf + 255) / 256);
    const int  blkMLP = (N + MT - 1) / MT;
    const int* src = ei;
    const int* dst = ei + E;

    gin_embed<<<blkNH, 256, 0, stream>>>(x, embW, embB, h, agg, totNH, Kin);

    for (int l = 0; l < L; ++l) {
        gin_zero<<<1, 256, 0, stream>>>(colsum, 2 * Hh);
        gin_scatter<<<blkE, 256, 0, stream>>>(h, src, dst, agg, totEf);
        gin_mlp<<<blkMLP, 256, 0, stream>>>(agg,
                                            W1 + (size_t)l * Hh * 2 * Hh,
                                            b1 + (size_t)l * 2 * Hh,
                                            W2 + (size_t)l * 2 * Hh * Hh,
                                            b2 + (size_t)l * Hh,
                                            colsum, colsq, (long)N);
        gin_bnstat<<<1, Hh, 0, stream>>>(colsum, colsq, gamma + (size_t)l * Hh,
                                         beta + (size_t)l * Hh, scale, shift,
                                         1.0f / (float)N);
        gin_apply<<<blkNH, 256, 0, stream>>>(scale, shift, agg, h, totNH);
    }
}